// ReasoningAwareAttention_87462714016309
// MI455X (gfx1250) — compile-verified
//
#include <hip/hip_runtime.h>

// Problem constants (from reference)
#define BB   4
#define HH   32
#define KVH  8
#define TT   1024
#define SS   1024
#define DD   128
#define HID  (HH * DD)          // 4096
#define KTOP 158
#define BOOST 2.5f
#define OUT_ELEMS ((long)BB * TT * HID)            // 16,777,216
#define PRUNED_ELEMS ((long)BB * HH * TT * SS)     // 134,217,728

typedef __attribute__((ext_vector_type(2))) float v2f;
typedef __attribute__((ext_vector_type(4))) float v4f;
typedef __attribute__((ext_vector_type(8))) float v8f;

// ---------------------------------------------------------------------------
// 1) Zero-fill the whole output (604 MB) with 128-bit non-temporal stores.
//    This is the bandwidth-critical kernel: ~26us floor at 23.3 TB/s.
// ---------------------------------------------------------------------------
__global__ void k_zero_fill(v4f* __restrict__ p, long n4, float* __restrict__ tail, long ntail) {
    long i = (long)blockIdx.x * blockDim.x + threadIdx.x;
    long stride = (long)gridDim.x * blockDim.x;
    v4f z = {0.0f, 0.0f, 0.0f, 0.0f};
    for (; i < n4; i += stride) {
        __builtin_nontemporal_store(z, p + i);
    }
    long t = (long)blockIdx.x * blockDim.x + threadIdx.x;
    if (t < ntail) tail[t] = 0.0f;
}

// ---------------------------------------------------------------------------
// 2) importance[b][s] = mean_h attn[b,h,T-1,s], boosted 2.5x at prompt ids
// ---------------------------------------------------------------------------
__global__ void k_importance(const float* __restrict__ attn,
                             const int* __restrict__ pids,
                             float* __restrict__ imp) {
    int b = blockIdx.x;
    int s = blockIdx.y * blockDim.x + threadIdx.x;   // 0..S-1
    const float* base = attn + (((long)b * HH) * TT + (TT - 1)) * SS + s;
    float acc = 0.0f;
    #pragma unroll
    for (int h = 0; h < HH; ++h) acc += base[(long)h * TT * SS];
    acc *= (1.0f / HH);
    bool inP = false;
    #pragma unroll
    for (int i = 0; i < 32; ++i) inP |= (pids[i] == s);
    imp[b * SS + s] = inP ? acc * BOOST : acc;
}

// ---------------------------------------------------------------------------
// 3) Rank-based top-K selection (matches jax.lax.top_k tie-break: lower index
//    wins on equal values). mask[b][s] = 1.0 iff s is among top-K of imp[b].
// ---------------------------------------------------------------------------
__global__ void k_topk_mask(const float* __restrict__ imp, float* __restrict__ mask) {
    int b = blockIdx.x;
    int s = threadIdx.x;                 // block = S = 1024 threads
    __shared__ float v[SS];
    v[s] = imp[b * SS + s];
    __syncthreads();
    float me = v[s];
    int cnt = 0;
    for (int j = 0; j < SS; ++j) {
        float x = v[j];
        cnt += (x > me) || (x == me && j < s);
    }
    mask[b * SS + s] = (cnt < KTOP) ? 1.0f : 0.0f;
}

// ---------------------------------------------------------------------------
// 4) pruned[b,h,T-1,s] = attn[b,h,T-1,s] * mask[b,s]   (only nonzero rows)
// ---------------------------------------------------------------------------
__global__ void k_pruned_lastrow(const float* __restrict__ attn,
                                 const float* __restrict__ mask,
                                 float* __restrict__ pruned) {
    long i = (long)blockIdx.x * blockDim.x + threadIdx.x;   // over B*H*S
    int  s  = (int)(i & (SS - 1));
    long bh = i >> 10;                 // b*H + h
    int  b  = (int)(bh >> 5);
    long off = (bh * TT + (TT - 1)) * SS + s;
    pruned[off] = attn[off] * mask[b * SS + s];
}

// ---------------------------------------------------------------------------
// 5) ctxPad[16 x HID]: rows 0..3 = sum_s pruned_row[b,h,s] * v[b,h/4,s,d],
//    rows 4..15 = 0 (M-padding for the 16x16 WMMA tile).
// ---------------------------------------------------------------------------
__global__ void k_ctx(const float* __restrict__ attn,
                      const float* __restrict__ mask,
                      const float* __restrict__ vkv,
                      float* __restrict__ ctxPad) {
    int h  = blockIdx.x;     // 0..31
    int bp = blockIdx.y;     // 0..15 (padded batch)
    int d  = threadIdx.x;    // 0..127
    if (bp >= BB) {          // zero pad rows
        ctxPad[(long)bp * HID + h * DD + d] = 0.0f;
        return;
    }
    __shared__ float w[SS];
    const float* arow = attn + (((long)(bp * HH + h)) * TT + (TT - 1)) * SS;
    const float* mrow = mask + bp * SS;
    for (int j = d; j < SS; j += blockDim.x) w[j] = arow[j] * mrow[j];
    __syncthreads();
    const float* vbase = vkv + ((long)(bp * KVH + (h / (HH / KVH))) * SS) * DD + d;
    float acc = 0.0f;
    for (int s = 0; s < SS; ++s) acc += w[s] * vbase[(long)s * DD];
    ctxPad[(long)bp * HID + h * DD + d] = acc;
}

// ---------------------------------------------------------------------------
// 6) out[b, T-1, :] = ctx[b,:] @ W^T via V_WMMA_F32_16X16X4_F32.
//    One wave per 16-wide N tile; K=4096 in steps of 4.
//    A layout (16x4 f32): lanes 0-15 -> M=lane, {VGPR0,VGPR1}={K0,K1};
//                         lanes 16-31 -> {K2,K3}.
//    B layout (4x16):     row K striped across lanes within a VGPR, mirrored.
//    C/D (16x16 f32):     VGPR r: lanes 0-15 = (M=r, N=lane); 16-31 = M=8+r.
// ---------------------------------------------------------------------------
__global__ void k_out_gemm_wmma(const float* __restrict__ ctxPad,
                                const float* __restrict__ W,
                                float* __restrict__ out) {
    int nt   = blockIdx.x;          // N tile: columns nt*16 .. nt*16+15
    int lane = threadIdx.x;         // 0..31 (one wave)
    int half = lane >> 4;           // 0 -> K pair (0,1), 1 -> K pair (2,3)
    int l    = lane & 15;

    const float* arow = ctxPad + (long)l * HID;             // A row M=l
    const float* brow = W + (long)(nt * 16 + l) * HID;      // B(k,n)=W[n][k]

    v8f c = {0.0f, 0.0f, 0.0f, 0.0f, 0.0f, 0.0f, 0.0f, 0.0f};
    #pragma unroll 4
    for (int k0 = 0; k0 < HID; k0 += 4) {
        v2f a, bv;
        a.x  = arow[k0 + 2 * half + 0];
        a.y  = arow[k0 + 2 * half + 1];
        bv.x = brow[k0 + 2 * half + 0];
        bv.y = brow[k0 + 2 * half + 1];
        c = __builtin_amdgcn_wmma_f32_16x16x4_f32(
                /*neg_a=*/false, a, /*neg_b=*/false, bv,
                /*c_mod=*/(short)0, c, /*reuse_a=*/false, /*reuse_b=*/false);
    }

    // Only M rows 0..3 are real batches; they live in lanes 0-15, VGPRs 0..3.
    if (half == 0) {
        #pragma unroll
        for (int r = 0; r < BB; ++r) {
            out[((long)(r * TT + (TT - 1))) * HID + nt * 16 + l] = c[r];
        }
    }
}

// ---------------------------------------------------------------------------
extern "C" void kernel_launch(void* const* d_in, const int* in_sizes, int n_in,
                              void* d_out, int out_size, void* d_ws, size_t ws_size,
                              hipStream_t stream) {
    const float* attn = (const float*)d_in[0];   // [B,H,T,S]
    const float* vkv  = (const float*)d_in[1];   // [B,KV,S,D]
    const float* W    = (const float*)d_in[2];   // [HID,HID]
    const int*   pids = (const int*)d_in[3];     // [32]

    float* out    = (float*)d_out;               // [B,T,HID]
    float* pruned = out + OUT_ELEMS;             // [B,H,T,S]

    float* imp    = (float*)d_ws;                // B*S
    float* mask   = imp + BB * SS;               // B*S
    float* ctxPad = mask + BB * SS;              // 16*HID

    // 1) zero-fill entire d_out (dominant cost; 128-bit NT stores)
    long total = (long)out_size;
    long n4    = total >> 2;
    long ntail = total - (n4 << 2);
    k_zero_fill<<<8192, 256, 0, stream>>>((v4f*)d_out, n4,
                                          (float*)d_out + (n4 << 2), ntail);

    // 2) importance
    k_importance<<<dim3(BB, SS / 256), 256, 0, stream>>>(attn, pids, imp);

    // 3) top-K mask
    k_topk_mask<<<BB, SS, 0, stream>>>(imp, mask);

    // 4) pruned last rows
    k_pruned_lastrow<<<(BB * HH * SS) / 256, 256, 0, stream>>>(attn, mask, pruned);

    // 5) context rows (M-padded to 16)
    k_ctx<<<dim3(HH, 16), DD, 0, stream>>>(attn, mask, vkv, ctxPad);

    // 6) output projection GEMM (WMMA f32)
    k_out_gemm_wmma<<<HID / 16, 32, 0, stream>>>(ctxPad, W, out);
}